// Resampling_4526895530285
// MI455X (gfx1250) — compile-verified
//
#include <hip/hip_runtime.h>
#include <stdint.h>

// MI455X / gfx1250, wave32. fp32 WMMA (16x16x4) GEMM fused with the
// patch-resampling gather. Memory-bound (~270 MB @ 23.3 TB/s ~ 12 us floor).
// W fragments are staged in LDS via async global->LDS copies (double-buffered)
// so each block reads packed-W from L2 exactly once (8x L2 traffic cut vs.
// per-wave streaming).

typedef __attribute__((ext_vector_type(2))) float v2f;
typedef __attribute__((ext_vector_type(4))) float v4f;
typedef __attribute__((ext_vector_type(8))) float v8f;
typedef __attribute__((ext_vector_type(4))) int   v4i;

#define GLB_AS __attribute__((address_space(1)))
#define LDS_AS __attribute__((address_space(3)))

#define BATCH 128
#define N1P   1024
#define D1C   256
#define NSTRIP (BATCH * N1P / 16)   // 8192 strips of 16 output rows
#define WPB    8                     // waves per block

#if __has_builtin(__builtin_amdgcn_global_load_async_to_lds_b128)
#define HAVE_ASYNC 1
#else
#define HAVE_ASYNC 0
#endif

#if __has_builtin(__builtin_amdgcn_s_wait_asynccnt)
#define ASYNC_WAIT(n) __builtin_amdgcn_s_wait_asynccnt(n)
#else
#define ASYNC_WAIT(n) asm volatile("s_wait_asynccnt %0" ::"i"(n) : "memory")
#endif

// Repack W[256][256] (row d, col e) into WMMA B-fragment order:
// t = ks*1024 + nt*64 + lane*2 + v  <-  W[ks*4 + 2*(lane/16) + v][nt*16 + lane%16]
__global__ __launch_bounds__(256) void repack_W_kernel(const float* __restrict__ W,
                                                       float* __restrict__ Wp) {
    int t    = blockIdx.x * blockDim.x + threadIdx.x;  // 0 .. 65535
    int v    = t & 1;
    int lane = (t >> 1) & 31;
    int nt   = (t >> 6) & 15;
    int ks   = t >> 10;
    int k = ks * 4 + 2 * (lane >> 4) + v;
    int e = nt * 16 + (lane & 15);
    Wp[t] = W[k * D1C + e];
}

__global__ __launch_bounds__(256) void resample_gemm_kernel(
    const float* __restrict__ enc,   // [128, 4096, 64]
    const float* __restrict__ Wp,    // packed W fragments, 64K floats
    const float* __restrict__ bias,  // [256]
    const float* __restrict__ pos,   // [1024, 256]
    float* __restrict__ out) {       // [131072, 256]
    __shared__ float ldsW[2][4096];  // double-buffered 16 KB W chunks (32 KB)

    const int tid   = threadIdx.x;
    const int lane  = tid & 31;
    const int wave  = tid >> 5;
    const int strip = blockIdx.x * WPB + wave;   // 0..8191
    const int m0    = strip << 4;
    const int h     = lane >> 4;                 // lane half -> K pair
    const int mr    = lane & 15;                 // A-operand M row

    // Per-thread cooperative staging addresses: chunk nt = Wp[ks*1024 + nt*64 + 0..63]
    // for ks = 0..63; thread t copies 4x16B: segment s covers ks = s*16 + t/16,
    // cols (t%16)*4 .. +3.  LDS float index f = s*1024 + t*4.
    const float* gW0 = Wp + (tid >> 4) * 1024 + ((tid & 15) << 2);
    float*       lW0 = &ldsW[0][tid << 2];

#if HAVE_ASYNC
    // Prefetch chunk 0 into buffer 0 (overlaps the A-panel gather below).
#pragma unroll
    for (int s = 0; s < 4; ++s)
        __builtin_amdgcn_global_load_async_to_lds_b128(
            (GLB_AS v4i*)(gW0 + s * 16384),
            (LDS_AS v4i*)(lW0 + s * 1024), 0, 0);
#endif

    // ---- Gather full K=256 A panel for this strip: 64 aligned b64 loads ----
    const int m   = m0 + mr;
    const int bi  = m >> 10;
    const int n1  = m & (N1P - 1);
    const int e1r = n1 >> 5;
    const int e1c = n1 & 31;
    const float* encb = enc + ((size_t)bi << 18);   // bi * 4096 * 64

    v2f af[64];
#pragma unroll
    for (int ks = 0; ks < 64; ++ks) {
        int d   = ks * 4 + 2 * h;                   // even -> contiguous pair
        int p1r = d >> 4;
        int p1c = d & 15;
        int y   = e1r * 16 + p1r;
        int x   = e1c * 16 + p1c;
        int idx = ((y >> 3) * 64 + (x >> 3)) * 64 + (y & 7) * 8 + (x & 7);
        af[ks] = *(const v2f*)(encb + idx);
    }

    // ---- 16 N-tiles; A panel reused from registers, W chunk from LDS ----
    for (int nt = 0; nt < 16; ++nt) {
#if HAVE_ASYNC
        const int buf = nt & 1;
        __syncthreads();   // all waves done reading buf^1 (iteration nt-1)
        if (nt < 15) {
#pragma unroll
            for (int s = 0; s < 4; ++s)
                __builtin_amdgcn_global_load_async_to_lds_b128(
                    (GLB_AS v4i*)(gW0 + (nt + 1) * 64 + s * 16384),
                    (LDS_AS v4i*)(lW0 + ((buf ^ 1) * 4096) + s * 1024), 0, 0);
            ASYNC_WAIT(4);  // in-order: <=4 outstanding => chunk nt landed
        } else {
            ASYNC_WAIT(0);
        }
        __syncthreads();    // chunk nt visible to all waves
        const float* lbuf = &ldsW[buf][0];
#else
        __syncthreads();
        v4f tmp[4];
#pragma unroll
        for (int s = 0; s < 4; ++s)
            tmp[s] = *(const v4f*)(gW0 + nt * 64 + s * 16384);
#pragma unroll
        for (int s = 0; s < 4; ++s)
            *(v4f*)(lW0 + s * 1024) = tmp[s];
        __syncthreads();
        const float* lbuf = &ldsW[0][0];
#endif

        v8f acc = {};
#pragma unroll
        for (int ks = 0; ks < 64; ++ks) {
            v2f wf = *(const v2f*)(lbuf + ks * 64 + lane * 2);  // ds_load_b64
            acc = __builtin_amdgcn_wmma_f32_16x16x4_f32(
                false, af[ks], false, wf, (short)0, acc, false, false);
        }

        // Epilogue: + bias[e] + pos_emb[n1, e]; C/D: vgpr r -> row r + 8h, col lane%16.
        int e  = nt * 16 + mr;
        float bv = bias[e];
#pragma unroll
        for (int r = 0; r < 8; ++r) {
            int mo  = m0 + r + 8 * h;
            int n1o = mo & (N1P - 1);
            out[((size_t)mo << 8) + e] = acc[r] + bv + pos[(n1o << 8) + e];
        }
    }
}

extern "C" void kernel_launch(void* const* d_in, const int* in_sizes, int n_in,
                              void* d_out, int out_size, void* d_ws, size_t ws_size,
                              hipStream_t stream) {
    const float* enc  = (const float*)d_in[0];   // encoded [128,4096,64]
    const float* W    = (const float*)d_in[1];   // [256,256]
    const float* bias = (const float*)d_in[2];   // [256]
    const float* pos  = (const float*)d_in[3];   // [1024,256]
    float* out = (float*)d_out;
    float* Wp  = (float*)d_ws;                   // 256 KB packed-W scratch

    repack_W_kernel<<<256, 256, 0, stream>>>(W, Wp);
    resample_gemm_kernel<<<NSTRIP / WPB, 256, 0, stream>>>(enc, Wp, bias, pos, out);
}